// BLayer_63780264346268
// MI455X (gfx1250) — compile-verified
//
#include <hip/hip_runtime.h>
#include <math.h>

// Problem constants (from reference)
#define OUTU   512
#define INW    1024
#define HID    32
#define BATCH  256
#define TOPK   6
#define NTHR   128      // 4 waves of 32
#define HS     34       // even padded stride: 8B-aligned K-pairs, conflict-free (34L mod 64 distinct)

typedef float v2f __attribute__((ext_vector_type(2)));
typedef float v8f __attribute__((ext_vector_type(8)));

// One 256x32 = relu( (256x32) @ (32x32) ) layer using V_WMMA_F32_16X16X4_F32.
// 32 output tiles of 16x16; wave `wid` owns tiles [wid*8, wid*8+8).
// A fragment (16x4 f32): lane L<16 -> row M=L, K=k0..k0+1 ; lane 16+L -> row M=L, K=k0+2..k0+3
// B fragment (4x16 f32): VGPR v: lanes 0-15 hold row K=k0+v (N=lane), lanes 16-31 row K=k0+2+v
// C/D (16x16 f32): VGPR r: M = r + (lane>>4)*8, N = lane&15
//
// src: LDS activations [b*HS + h], HS even -> v2f loads are one ds_load_b64 each.
// wp : LDS weights pre-swizzled as [(k>>1)*64 + n*2 + (k&1)] -> B pair contiguous.
__device__ __forceinline__ void layer_wmma(const float* __restrict__ src,
                                           const float* __restrict__ wp,
                                           float* __restrict__ dst,
                                           int wid, int l15, int kh)
{
    for (int tt = 0; tt < 8; ++tt) {
        const int tile = wid * 8 + tt;
        const int m0 = (tile >> 1) * 16;
        const int n0 = (tile & 1) * 16;
        v8f c = {};
#pragma unroll
        for (int k0 = 0; k0 < HID; k0 += 4) {
            const int ka = k0 + kh * 2;                       // even
            v2f a = *(const v2f*)&src[(m0 + l15) * HS + ka];  // rows K=ka, ka+1 of A^T pair
            v2f b = *(const v2f*)&wp[(ka >> 1) * 64 + (n0 + l15) * 2];
            c = __builtin_amdgcn_wmma_f32_16x16x4_f32(false, a, false, b,
                                                      (short)0, c, false, false);
        }
#pragma unroll
        for (int r = 0; r < 8; ++r) {
            // single v_max_num_f32 (llvm.maxnum), avoids canonicalize+max pair
            dst[(m0 + r + kh * 8) * HS + n0 + l15] = __builtin_fmaxf(c[r], 0.0f);
        }
    }
}

// Stage a 32x32 row-major weight from global into K-pair-interleaved LDS layout.
__device__ __forceinline__ void stage_weight_swizzled(const float* __restrict__ g,
                                                      float* __restrict__ wp, int t)
{
    for (int p = t; p < HID * HID; p += NTHR) {
        int k = p >> 5, n = p & 31;
        wp[(k >> 1) * 64 + n * 2 + (k & 1)] = g[p];
    }
}

__global__ __launch_bounds__(NTHR, 1)
void blayer_topk_mlp_kernel(const float* __restrict__ inputs,  // (B, IN)
                            const float* __restrict__ mask,    // (OUT, IN)
                            const float* __restrict__ W1,      // (OUT, IN, HID)
                            const float* __restrict__ W2,      // (OUT, HID, HID)
                            const float* __restrict__ W3,      // (OUT, HID, HID)
                            const float* __restrict__ W4,      // (OUT, HID, 1)
                            float* __restrict__ out)           // (B, OUT)
{
    __shared__ float s_mask[INW];
    __shared__ float s_rv[NTHR];
    __shared__ int   s_ri[NTHR];
    __shared__ int   s_idx[8];
    __shared__ float s_in[TOPK * BATCH];               // [j][b] gathered input columns
    __shared__ float s_w1[TOPK * HID];                 // [j][h] gathered W1 rows
    __shared__ alignas(16) float s_wp[HID * HID];      // swizzled dense weight
    __shared__ float s_w4[HID];
    __shared__ alignas(16) float s_h0[BATCH * HS];
    __shared__ alignas(16) float s_h1[BATCH * HS];

    const int o    = blockIdx.x;
    const int t    = threadIdx.x;
    const int lane = t & 31;
    const int wid  = t >> 5;
    const int l15  = lane & 15;
    const int kh   = lane >> 4;

    // ---- Phase 1: load mask row ----
    {
        const float4* src = (const float4*)(mask + (size_t)o * INW);
        float4* dst = (float4*)s_mask;
        for (int p = t; p < INW / 4; p += NTHR) dst[p] = src[p];
    }
    __syncthreads();

    // ---- Phase 2: top-6 via iterative block argmax (ties -> lower index) ----
    for (int r = 0; r < TOPK; ++r) {
        float bv = -INFINITY; int bi = 0x7fffffff;
        for (int p = t; p < INW; p += NTHR) {
            float v = s_mask[p];
            if (v > bv) { bv = v; bi = p; }   // strided ascending p => first max kept
        }
        s_rv[t] = bv; s_ri[t] = bi;
        __syncthreads();
        for (int off = NTHR / 2; off > 0; off >>= 1) {
            if (t < off) {
                float ov = s_rv[t + off]; int oi = s_ri[t + off];
                float mv = s_rv[t];       int mi = s_ri[t];
                if (ov > mv || (ov == mv && oi < mi)) { s_rv[t] = ov; s_ri[t] = oi; }
            }
            __syncthreads();
        }
        if (t == 0) { int w = s_ri[0]; s_idx[r] = w; s_mask[w] = -INFINITY; }
        __syncthreads();
    }

    // ---- Phase 3: gather 6 input columns + 6 W1 rows; stage W2 (swizzled) + W4 ----
    for (int p = t; p < TOPK * BATCH; p += NTHR) {
        int j = p >> 8;             // p / 256
        int b = p & 255;
        s_in[p] = inputs[(size_t)b * INW + s_idx[j]];
    }
    for (int p = t; p < TOPK * HID; p += NTHR) {
        int j = p >> 5, h = p & 31;
        s_w1[p] = W1[(size_t)o * INW * HID + (size_t)s_idx[j] * HID + h];
    }
    stage_weight_swizzled(W2 + (size_t)o * HID * HID, s_wp, t);
    if (t < HID) s_w4[t] = W4[(size_t)o * HID + t];
    __syncthreads();

    // ---- Phase 4: layer 1 (sparse K=6) on VALU ----
    for (int p = t; p < BATCH * HID; p += NTHR) {
        int b = p >> 5, h = p & 31;
        float acc = 0.f;
#pragma unroll
        for (int j = 0; j < TOPK; ++j)
            acc = fmaf(s_in[j * BATCH + b], s_w1[j * HID + h], acc);
        s_h0[b * HS + h] = __builtin_fmaxf(acc, 0.0f);
    }
    __syncthreads();

    // ---- Phase 5: layer 2 (WMMA f32), h0 -> h1 ----
    layer_wmma(s_h0, s_wp, s_h1, wid, l15, kh);
    __syncthreads();

    // ---- Phase 6: swap in W3 (swizzled) ----
    stage_weight_swizzled(W3 + (size_t)o * HID * HID, s_wp, t);
    __syncthreads();

    // ---- Phase 7: layer 3 (WMMA f32), h1 -> h0 ----
    layer_wmma(s_h1, s_wp, s_h0, wid, l15, kh);
    __syncthreads();

    // ---- Phase 8: layer 4 dot + sigmoid-threshold binarize ----
    // sigmoid(z) >= 0.5  <=>  z >= 0, so sigmoid is elided.
    for (int b = t; b < BATCH; b += NTHR) {
        float acc = 0.f;
#pragma unroll
        for (int h = 0; h < HID; ++h)
            acc = fmaf(s_h0[b * HS + h], s_w4[h], acc);
        out[(size_t)b * OUTU + o] = (acc >= 0.f) ? 1.0f : -1.0f;
    }
}

extern "C" void kernel_launch(void* const* d_in, const int* in_sizes, int n_in,
                              void* d_out, int out_size, void* d_ws, size_t ws_size,
                              hipStream_t stream) {
    (void)in_sizes; (void)n_in; (void)out_size; (void)d_ws; (void)ws_size;
    const float* inputs = (const float*)d_in[0];
    const float* mask   = (const float*)d_in[1];
    const float* W1     = (const float*)d_in[2];
    const float* W2     = (const float*)d_in[3];
    const float* W3     = (const float*)d_in[4];
    const float* W4     = (const float*)d_in[5];
    float* out = (float*)d_out;
    blayer_topk_mlp_kernel<<<OUTU, NTHR, 0, stream>>>(inputs, mask, W1, W2, W3, W4, out);
}